// CoAttentionLayer_618475291110
// MI455X (gfx1250) — compile-verified
//
#include <hip/hip_runtime.h>
#include <math.h>

// ---------------------------------------------------------------------------
// CoAttention scores on MI455X (gfx1250, wave32)
//   B=64, R=A=128, F=256, D=128
//   Phase 1: keys = receiver @ w_k ; queries' = attendant @ w_q + bias
//            -> V_WMMA_F32_16X16X4_F32 (f32 matrix path, K stepped by 4)
//   Phase 2: out[b,r,a] = sum_d tanh(q'[b,a,d] + k[b,r,d]) * a[d]
//            -> LDS-staged, conflict-free transposed q, native V_TANH_F32,
//               2r x 4a register blocking (0.875 LDS dwords / output elem)
// ---------------------------------------------------------------------------

#define Bsz 64
#define Rsz 128
#define Asz 128
#define Fsz 256
#define Dsz 128

typedef __attribute__((ext_vector_type(2))) float v2f;
typedef __attribute__((ext_vector_type(8))) float v8f;

// Native TRANS tanh on gfx1250 (V_TANH_F32); OCML fallback otherwise.
#if __has_builtin(__builtin_amdgcn_tanhf)
  #define TANH(x) __builtin_amdgcn_tanhf(x)
#elif __has_builtin(__builtin_amdgcn_tanh_f32)
  #define TANH(x) __builtin_amdgcn_tanh_f32(x)
#else
  #define TANH(x) tanhf(x)
#endif

// ---------------------------------------------------------------------------
// Phase 1: 16x16 output tile per wave, K=256 in steps of 4 (64 WMMAs/tile).
// Tile space: 2 matrices x 512 M-tiles x 8 N-tiles = 8192 tiles.
// 1024 blocks x 8 waves.
//
// f32 WMMA operand layout (ISA 7.12.2):
//   A 16x4 : lanes 0-15 -> M=lane, VGPR0=K0,VGPR1=K1 ; lanes 16-31 -> K2,K3
//   B 4x16 : lanes 0-15 -> N=lane, VGPR0=K0,VGPR1=K1 ; lanes 16-31 -> K2,K3
//   C/D    : VGPR v -> M = v + (lane>=16 ? 8 : 0), N = lane&15
// ---------------------------------------------------------------------------
__global__ __launch_bounds__(256) void coattn_proj_wmma(
    const float* __restrict__ receiver,   // [B*R, F]
    const float* __restrict__ attendant,  // [B*A, F]
    const float* __restrict__ w_k,        // [F, D]
    const float* __restrict__ w_q,        // [F, D]
    const float* __restrict__ bias,       // [D]
    float* __restrict__ keys_ws,          // [B*R, D]
    float* __restrict__ qry_ws)           // [B*A, D]  (bias folded in)
{
    const int t    = threadIdx.x;
    const int lane = t & 31;
    const int wave = t >> 5;

    const int tile  = blockIdx.x * 8 + wave;   // 0..8191
    const int which = tile >> 12;              // 0 = keys, 1 = queries
    const int tt    = tile & 4095;
    const int mt    = tt >> 3;                 // 0..511 (M tile)
    const int nt    = tt & 7;                  // 0..7   (N tile)

    const float* X = which ? attendant : receiver;
    const float* W = which ? w_q : w_k;
    float*       O = which ? qry_ws : keys_ws;

    const int mrow = mt * 16 + (lane & 15);
    const int ncol = nt * 16 + (lane & 15);
    const int koff = (lane >> 4) << 1;         // 0 for lanes 0-15, 2 for 16-31

    const float* arow = X + (size_t)mrow * Fsz;

    v8f c = {};
    #pragma unroll 4
    for (int k0 = 0; k0 < Fsz; k0 += 4) {
        v2f a, b;
        a.x = arow[k0 + koff];
        a.y = arow[k0 + koff + 1];
        b.x = W[(size_t)(k0 + koff) * Dsz + ncol];
        b.y = W[(size_t)(k0 + koff + 1) * Dsz + ncol];
        // 8 args: (neg_a, A, neg_b, B, c_mod, C, reuse_a, reuse_b)
        c = __builtin_amdgcn_wmma_f32_16x16x4_f32(
                false, a, false, b, (short)0, c, false, false);
    }

    const float badd  = which ? bias[ncol] : 0.0f;
    const int   rbase = mt * 16 + (lane >> 4) * 8;
    #pragma unroll
    for (int v = 0; v < 8; ++v) {
        O[(size_t)(rbase + v) * Dsz + ncol] = c[v] + badd;
    }
}

// ---------------------------------------------------------------------------
// Phase 2: one block = one (batch, 32-row r-slab). 512 threads = 16 waves.
// Wave w owns r = rblk*32 + 2w + {0,1}; lanes sweep a in 4 chunks of 32.
// 8 accumulators/lane -> per d-step: 4 q loads + 2 k bcast + 1 a bcast
// = 7 LDS reads for 8 output contributions.
// LDS: q_t[d][a] with row stride 129 dwords (conflict-free both directions:
// transposing store has lane-stride 129 -> bank step 1; compute load has
// lane-stride 1), k slab [32][128], a vector [128]. Total 82,944 B.
// ---------------------------------------------------------------------------
#define QT_STRIDE 129

__global__ __launch_bounds__(512) void coattn_score(
    const float* __restrict__ qry_ws,   // [B*A, D], bias folded
    const float* __restrict__ keys_ws,  // [B*R, D]
    const float* __restrict__ avec,     // [D]
    float* __restrict__ out)            // [B, R, A]
{
    extern __shared__ float smem[];
    float* q_t = smem;                       // 128 * 129
    float* kl  = smem + Dsz * QT_STRIDE;     // 32 * 128
    float* al  = kl + 32 * Dsz;              // 128

    const int b    = blockIdx.x >> 2;        // /4
    const int rblk = blockIdx.x & 3;         // %4  (slab of 32 rows)
    const int t    = threadIdx.x;            // 0..511
    const int lane = t & 31;
    const int wave = t >> 5;                 // 0..15

    // Transposing fill of q_t: consecutive threads -> consecutive d.
    const float* qb = qry_ws + (size_t)b * Asz * Dsz;
    #pragma unroll
    for (int i = 0; i < (Asz * Dsz) / 512; ++i) {
        int idx = t + i * 512;
        int a   = idx >> 7;
        int d   = idx & 127;
        q_t[d * QT_STRIDE + a] = qb[idx];
    }
    // k slab (32 rows), coalesced + conflict-free
    const float* kb = keys_ws + (size_t)(b * Rsz + rblk * 32) * Dsz;
    #pragma unroll
    for (int i = 0; i < (32 * Dsz) / 512; ++i) {
        int idx = t + i * 512;
        kl[idx] = kb[idx];
    }
    if (t < Dsz) al[t] = avec[t];
    __syncthreads();

    const float* k0row = kl + (wave * 2 + 0) * Dsz;
    const float* k1row = kl + (wave * 2 + 1) * Dsz;

    float a00 = 0.f, a01 = 0.f, a02 = 0.f, a03 = 0.f;   // r0, a-chunks 0..3
    float a10 = 0.f, a11 = 0.f, a12 = 0.f, a13 = 0.f;   // r1, a-chunks 0..3

    #pragma unroll 4
    for (int d = 0; d < Dsz; ++d) {
        const float kv0 = k0row[d];               // wave-uniform broadcast
        const float kv1 = k1row[d];               // wave-uniform broadcast
        const float av  = al[d];                  // wave-uniform broadcast
        const float* qrow = q_t + d * QT_STRIDE + lane;
        const float q0 = qrow[0];
        const float q1 = qrow[32];
        const float q2 = qrow[64];
        const float q3 = qrow[96];
        a00 += TANH(q0 + kv0) * av;               // V_TANH_F32 TRANS ops,
        a01 += TANH(q1 + kv0) * av;               // co-execute with VALU fma
        a02 += TANH(q2 + kv0) * av;
        a03 += TANH(q3 + kv0) * av;
        a10 += TANH(q0 + kv1) * av;
        a11 += TANH(q1 + kv1) * av;
        a12 += TANH(q2 + kv1) * av;
        a13 += TANH(q3 + kv1) * av;
    }

    const int r0 = rblk * 32 + wave * 2;
    float* orow0 = out + ((size_t)(b * Rsz + r0)) * Asz + lane;
    float* orow1 = orow0 + Asz;
    orow0[0]  = a00;  orow0[32] = a01;  orow0[64] = a02;  orow0[96] = a03;
    orow1[0]  = a10;  orow1[32] = a11;  orow1[64] = a12;  orow1[96] = a13;
}

// ---------------------------------------------------------------------------
extern "C" void kernel_launch(void* const* d_in, const int* in_sizes, int n_in,
                              void* d_out, int out_size, void* d_ws, size_t ws_size,
                              hipStream_t stream) {
    const float* receiver  = (const float*)d_in[0];  // [B,R,F]
    const float* attendant = (const float*)d_in[1];  // [B,A,F]
    const float* w_q       = (const float*)d_in[2];  // [F,D]
    const float* w_k       = (const float*)d_in[3];  // [F,D]
    const float* bias      = (const float*)d_in[4];  // [D]
    const float* avec      = (const float*)d_in[5];  // [D]
    float*       out       = (float*)d_out;          // [B,R,A]

    float* qry_ws  = (float*)d_ws;                        // 4 MB
    float* keys_ws = qry_ws + (size_t)Bsz * Asz * Dsz;    // 4 MB

    // Phase 1: 8192 16x16 tiles, 8 waves/block -> 1024 blocks of 256 threads.
    coattn_proj_wmma<<<1024, 256, 0, stream>>>(
        receiver, attendant, w_k, w_q, bias, keys_ws, qry_ws);

    // Phase 2: 64 batches x 4 r-slabs = 256 blocks of 512 threads.
    const size_t lds_bytes = (Dsz * QT_STRIDE + 32 * Dsz + Dsz) * sizeof(float);
    coattn_score<<<256, 512, lds_bytes, stream>>>(
        qry_ws, keys_ws, avec, out);
}